// RGCN_73254962201301
// MI455X (gfx1250) — compile-verified
//
#include <hip/hip_runtime.h>

// RGCN forward for MI455X (gfx1250, wave32).
//  - fp32 WMMA (V_WMMA_F32_16X16X4_F32) for all dense GEMMs (exact fp32).
//  - CSR-gather aggregation (count -> LDS scan -> fill), one wave per dst row:
//    pure L2-read bound (all 51.2MB tables resident in 192MB L2), zero float
//    atomics on the hot path (vs ~820M in a scatter formulation).
//  - Weights pre-swizzled into WMMA fragment order so B loads are b64.
//  - Bias/ReLU/self-loop fused into gather / GEMM epilogues.

typedef __attribute__((ext_vector_type(2))) float v2f;
typedef __attribute__((ext_vector_type(8))) float v8f;

#define IN_DIM  128
#define HID_DIM 128
#define OUT_DIM 64

// ---------------- tiny utility kernels ----------------

__global__ void rgcn_zero_i32_kernel(int* __restrict__ p, int n) {
  int i = blockIdx.x * blockDim.x + threadIdx.x;
  if (i < n) p[i] = 0;
}

__global__ void rgcn_bias_sum_kernel(float* __restrict__ o, const float* __restrict__ a,
                                     const float* __restrict__ b, const float* __restrict__ c,
                                     int n) {
  int i = blockIdx.x * blockDim.x + threadIdx.x;
  if (i < n) {
    float v = a[i];
    if (b) v += b[i];
    if (c) v += c[i];
    o[i] = v;
  }
}

// ---------------- CSR build: count, exclusive scan, fill ----------------

__global__ void rgcn_count_kernel(const int* __restrict__ dst, int* __restrict__ cnt, int nE) {
  int e = blockIdx.x * blockDim.x + threadIdx.x;
  if (e < nE) atomicAdd(&cnt[dst[e]], 1);
}

// Per-block exclusive scan (Hillis-Steele in LDS) + block totals.
__global__ void rgcn_scan1_kernel(const int* __restrict__ in, int* __restrict__ out,
                                  int* __restrict__ bsum, int n) {
  __shared__ int buf[256];
  const int t = threadIdx.x;
  const int i = blockIdx.x * 256 + t;
  const int v = (i < n) ? in[i] : 0;
  buf[t] = v;
  __syncthreads();
  for (int off = 1; off < 256; off <<= 1) {
    int x = (t >= off) ? buf[t - off] : 0;
    __syncthreads();
    buf[t] += x;
    __syncthreads();
  }
  if (i < n) out[i] = buf[t] - v;          // exclusive
  if (t == 255) bsum[blockIdx.x] = buf[255];
}

// Single-block exclusive scan of block totals (nb <= 1024).
__global__ void rgcn_scan2_kernel(int* __restrict__ bsum, int nb) {
  __shared__ int buf[1024];
  const int t = threadIdx.x;
  const int v = (t < nb) ? bsum[t] : 0;
  buf[t] = v;
  __syncthreads();
  for (int off = 1; off < 1024; off <<= 1) {
    int x = (t >= off) ? buf[t - off] : 0;
    __syncthreads();
    buf[t] += x;
    __syncthreads();
  }
  if (t < nb) bsum[t] = buf[t] - v;        // exclusive
}

__global__ void rgcn_scan3_kernel(int* __restrict__ out, const int* __restrict__ bsum, int n) {
  int i = blockIdx.x * 256 + threadIdx.x;
  if (i < n) out[i] += bsum[blockIdx.x];
}

__global__ void rgcn_fill_kernel(const int* __restrict__ src, const int* __restrict__ dst,
                                 const int* __restrict__ rowptr, int* __restrict__ cur,
                                 int* __restrict__ col, int nE) {
  int e = blockIdx.x * blockDim.x + threadIdx.x;
  if (e < nE) {
    int d = dst[e];
    int p = atomicAdd(&cur[d], 1);
    col[rowptr[d] + p] = src[e];
  }
}

// ---------------- weight pre-swizzle into WMMA fragment order ----------------
// P[f*64 + lane*2 + c] = W[(k0 + 2*(lane>>4) + c) * N + t*16 + (lane&15)],
// where f = (k0/4)*NT + t. Each lane's B fragment becomes one b64 load.
template<int N, int K>
__global__ void rgcn_repack_w_kernel(const float* __restrict__ W, float* __restrict__ P) {
  constexpr int NT = N / 16;
  constexpr int NF = (K / 4) * NT;
  int idx = blockIdx.x * blockDim.x + threadIdx.x;   // f*32 + lane
  if (idx >= NF * 32) return;
  const int f = idx >> 5, l = idx & 31;
  const int k0 = (f / NT) * 4, t = f % NT;
  const int krow = k0 + 2 * (l >> 4);
  const int ncol = t * 16 + (l & 15);
  P[idx * 2 + 0] = W[(size_t)(krow + 0) * N + ncol];
  P[idx * 2 + 1] = W[(size_t)(krow + 1) * N + ncol];
}

// ---------------- fp32 WMMA GEMM: C[M,N] = A[M,K] @ W (+bias) ----------------
// One wave owns a 16-row strip x all N columns (N/16 v8f accumulators); A frag
// reused across N tiles; B frags from the pre-swizzled table (b64 per lane).
template<int N, int K>
__global__ void rgcn_gemm_wmma_kernel(const float* __restrict__ A,
                                      const float* __restrict__ Pw,
                                      const float* __restrict__ bias,
                                      float* __restrict__ C, int M) {
  const int wave = (int)((blockIdx.x * (size_t)blockDim.x + threadIdx.x) >> 5);
  const int lane = threadIdx.x & 31;
  const int row0 = wave << 4;
  if (row0 >= M) return;                 // whole-wave exit only (EXEC all-1 for WMMA)
  const int half = lane >> 4;
  const int l16  = lane & 15;
  const int kk   = half * 2;

  int arow = row0 + l16;
  if (arow >= M) arow = M - 1;           // clamp (M is a multiple of 16 here)
  const float* Ap = A + (size_t)arow * K;

  constexpr int NT = N / 16;
  v8f acc[NT] = {};

  #pragma unroll 4
  for (int k0 = 0; k0 < K; k0 += 4) {
    v2f a;
    a.x = Ap[k0 + kk + 0];
    a.y = Ap[k0 + kk + 1];
    const float* Pb = Pw + (size_t)(k0 >> 2) * NT * 64 + lane * 2;
    #pragma unroll
    for (int t = 0; t < NT; ++t) {
      v2f b = *(const v2f*)(Pb + t * 64);
      acc[t] = __builtin_amdgcn_wmma_f32_16x16x4_f32(
          /*neg_a=*/false, a, /*neg_b=*/false, b,
          /*c_mod=*/(short)0, acc[t], /*reuse_a=*/false, /*reuse_b=*/false);
    }
  }

  const int srow = row0 + half * 8;
  #pragma unroll
  for (int t = 0; t < NT; ++t) {
    const float bv = bias ? bias[t * 16 + l16] : 0.0f;
    #pragma unroll
    for (int v = 0; v < 8; ++v) {
      int r = srow + v;
      if (r < M) C[(size_t)r * N + t * 16 + l16] = acc[t][v] + bv;
    }
  }
}

// ---------------- CSR gather (segment mean-ish: sum * 1/max(deg,1)) ----------------

template<int PER>
__device__ __forceinline__ void rgcn_row_sum(const float* __restrict__ msg,
                                             const int* __restrict__ col,
                                             int start, int end, int lane,
                                             float* __restrict__ acc) {
  constexpr int W = PER * 32;
  #pragma unroll
  for (int c = 0; c < PER; ++c) acc[c] = 0.0f;
  int j = start;
  for (; j + 4 <= end; j += 4) {
    const int c0 = col[j], c1 = col[j + 1], c2 = col[j + 2], c3 = col[j + 3];
    if (j + 8 <= end) {  // prefetch next chunk's rows (global_prefetch_b8)
      __builtin_prefetch(msg + (size_t)col[j + 4] * W + lane * PER, 0, 0);
      __builtin_prefetch(msg + (size_t)col[j + 5] * W + lane * PER, 0, 0);
      __builtin_prefetch(msg + (size_t)col[j + 6] * W + lane * PER, 0, 0);
      __builtin_prefetch(msg + (size_t)col[j + 7] * W + lane * PER, 0, 0);
    }
    const float* r0 = msg + (size_t)c0 * W + lane * PER;
    const float* r1 = msg + (size_t)c1 * W + lane * PER;
    const float* r2 = msg + (size_t)c2 * W + lane * PER;
    const float* r3 = msg + (size_t)c3 * W + lane * PER;
    if constexpr (PER == 4) {
      float4 v0 = *(const float4*)r0, v1 = *(const float4*)r1;
      float4 v2 = *(const float4*)r2, v3 = *(const float4*)r3;
      acc[0] += v0.x + v1.x + v2.x + v3.x;
      acc[1] += v0.y + v1.y + v2.y + v3.y;
      acc[2] += v0.z + v1.z + v2.z + v3.z;
      acc[3] += v0.w + v1.w + v2.w + v3.w;
    } else {
      float2 v0 = *(const float2*)r0, v1 = *(const float2*)r1;
      float2 v2 = *(const float2*)r2, v3 = *(const float2*)r3;
      acc[0] += v0.x + v1.x + v2.x + v3.x;
      acc[1] += v0.y + v1.y + v2.y + v3.y;
    }
  }
  for (; j < end; ++j) {
    const float* r = msg + (size_t)col[j] * W + lane * PER;
    if constexpr (PER == 4) {
      float4 v = *(const float4*)r;
      acc[0] += v.x; acc[1] += v.y; acc[2] += v.z; acc[3] += v.w;
    } else {
      float2 v = *(const float2*)r;
      acc[0] += v.x; acc[1] += v.y;
    }
  }
}

// One wave per destination row. Optionally a second edge type (TWO), optional
// ReLU, optional accumulate-onto-existing-output (ACC). Non-atomic row write.
template<int W, bool TWO, bool RELU, bool ACC>
__global__ void rgcn_gather_kernel(const float* __restrict__ msgA, const int* __restrict__ rpA,
                                   const int* __restrict__ cntA, const int* __restrict__ colA,
                                   const float* __restrict__ msgB, const int* __restrict__ rpB,
                                   const int* __restrict__ cntB, const int* __restrict__ colB,
                                   const float* __restrict__ bias,
                                   float* __restrict__ out, int nDst) {
  constexpr int PER = W / 32;
  const int lane = threadIdx.x & 31;
  const int d = (int)((blockIdx.x * (size_t)blockDim.x + threadIdx.x) >> 5);
  if (d >= nDst) return;

  const int sA = rpA[d], cA = cntA[d];
  float accA[PER];
  rgcn_row_sum<PER>(msgA, colA, sA, sA + cA, lane, accA);
  const float normA = 1.0f / fmaxf((float)cA, 1.0f);

  float res[PER];
  #pragma unroll
  for (int c = 0; c < PER; ++c) res[c] = accA[c] * normA;

  if constexpr (TWO) {
    const int sB = rpB[d], cB = cntB[d];
    float accB[PER];
    rgcn_row_sum<PER>(msgB, colB, sB, sB + cB, lane, accB);
    const float normB = 1.0f / fmaxf((float)cB, 1.0f);
    #pragma unroll
    for (int c = 0; c < PER; ++c) res[c] += accB[c] * normB;
  }

  if (bias) {
    #pragma unroll
    for (int c = 0; c < PER; ++c) res[c] += bias[lane * PER + c];
  }

  float* orow = out + (size_t)d * W + lane * PER;
  if constexpr (ACC) {
    if constexpr (PER == 4) {
      float4 o = *(const float4*)orow;
      res[0] += o.x; res[1] += o.y; res[2] += o.z; res[3] += o.w;
    } else {
      float2 o = *(const float2*)orow;
      res[0] += o.x; res[1] += o.y;
    }
  }
  if constexpr (RELU) {
    #pragma unroll
    for (int c = 0; c < PER; ++c) res[c] = fmaxf(res[c], 0.0f);
  }
  if constexpr (PER == 4) {
    *(float4*)orow = make_float4(res[0], res[1], res[2], res[3]);
  } else {
    *(float2*)orow = make_float2(res[0], res[1]);
  }
}

// ---------------- host-side orchestration ----------------

extern "C" void kernel_launch(void* const* d_in, const int* in_sizes, int n_in,
                              void* d_out, int out_size, void* d_ws, size_t ws_size,
                              hipStream_t stream) {
  (void)n_in; (void)out_size; (void)ws_size;
  const float* x_user      = (const float*)d_in[0];
  const float* x_item      = (const float*)d_in[1];
  const int*   rates_src   = (const int*)d_in[2];
  const int*   rates_dst   = (const int*)d_in[3];
  const int*   ratedby_src = (const int*)d_in[4];
  const int*   ratedby_dst = (const int*)d_in[5];
  const int*   sim_src     = (const int*)d_in[6];
  const int*   sim_dst     = (const int*)d_in[7];
  const float* W0_rates    = (const float*)d_in[8];
  const float* b0_rates    = (const float*)d_in[9];
  const float* W0_ratedby  = (const float*)d_in[10];
  const float* b0_ratedby  = (const float*)d_in[11];
  const float* W0_sim      = (const float*)d_in[12];
  const float* b0_sim      = (const float*)d_in[13];
  const float* W1_rates    = (const float*)d_in[14];
  const float* b1_rates    = (const float*)d_in[15];
  const float* W1_sim      = (const float*)d_in[16];
  const float* b1_sim      = (const float*)d_in[17];
  const float* W_self      = (const float*)d_in[18];
  const float* b_self      = (const float*)d_in[19];

  const int Mu      = in_sizes[0] / IN_DIM;
  const int Mi      = in_sizes[1] / IN_DIM;
  const int E_rates = in_sizes[2];
  const int E_rb    = in_sizes[4];
  const int E_sim   = in_sizes[6];

  // ---- workspace bump allocator (256B aligned blocks) ----
  char* wp = (char*)d_ws;
  auto alloc_f = [&](size_t n) { float* r = (float*)wp; wp += ((n * 4 + 255) / 256) * 256; return r; };
  auto alloc_i = [&](size_t n) { int*   r = (int*)wp;   wp += ((n * 4 + 255) / 256) * 256; return r; };

  const size_t SU = (size_t)Mu * HID_DIM;
  const size_t SI = (size_t)Mi * HID_DIM;
  float* tmp_user     = alloc_f(SU);     // x_user @ W0_rates
  float* tmp_item_sim = alloc_f(SI);     // x_item @ W0_sim
  float* tmp_item_rb  = alloc_f(SI);     // x_item @ W0_ratedby
  float* h_item       = alloc_f(SI);
  float* h_user       = alloc_f(SU);
  float* bias_hi      = alloc_f(HID_DIM);   // b0_rates + b0_sim
  float* out_bias     = alloc_f(OUT_DIM);   // b_self + b1_rates + b1_sim
  float* P0_rates     = alloc_f((IN_DIM / 4) * (HID_DIM / 16) * 64);
  float* P0_sim       = alloc_f((IN_DIM / 4) * (HID_DIM / 16) * 64);
  float* P0_rb        = alloc_f((IN_DIM / 4) * (HID_DIM / 16) * 64);
  float* P1_rates     = alloc_f((HID_DIM / 4) * (OUT_DIM / 16) * 64);
  float* P1_sim       = alloc_f((HID_DIM / 4) * (OUT_DIM / 16) * 64);
  float* P_self       = alloc_f((HID_DIM / 4) * (OUT_DIM / 16) * 64);
  // CSR per edge type (dst-indexed)
  int* rpA  = alloc_i(Mi); int* cntA = alloc_i(Mi); int* curA = alloc_i(Mi); int* colA = alloc_i(E_rates); // rates->item
  int* rpB  = alloc_i(Mi); int* cntB = alloc_i(Mi); int* curB = alloc_i(Mi); int* colB = alloc_i(E_sim);   // sim->item
  int* rpC  = alloc_i(Mu); int* cntC = alloc_i(Mu); int* curC = alloc_i(Mu); int* colC = alloc_i(E_rb);    // ratedby->user
  int* bsum = alloc_i(1024);
  float* out = (float*)d_out;            // [Mi, 64]
  float* tmp2_user = tmp_user;           // alias: h_user @ W1_rates  [Mu,64]
  float* tmp2_item = tmp_item_sim;       // alias: h_item @ W1_sim    [Mi,64]

  const int B = 256;
  auto cdiv = [](long a, long b) { return (int)((a + b - 1) / b); };

  // ---- CSR build (3 edge types) ----
  rgcn_zero_i32_kernel<<<cdiv(Mi, B), B, 0, stream>>>(cntA, Mi);
  rgcn_zero_i32_kernel<<<cdiv(Mi, B), B, 0, stream>>>(curA, Mi);
  rgcn_zero_i32_kernel<<<cdiv(Mi, B), B, 0, stream>>>(cntB, Mi);
  rgcn_zero_i32_kernel<<<cdiv(Mi, B), B, 0, stream>>>(curB, Mi);
  rgcn_zero_i32_kernel<<<cdiv(Mu, B), B, 0, stream>>>(cntC, Mu);
  rgcn_zero_i32_kernel<<<cdiv(Mu, B), B, 0, stream>>>(curC, Mu);
  rgcn_count_kernel<<<cdiv(E_rates, B), B, 0, stream>>>(rates_dst,   cntA, E_rates);
  rgcn_count_kernel<<<cdiv(E_sim,   B), B, 0, stream>>>(sim_dst,     cntB, E_sim);
  rgcn_count_kernel<<<cdiv(E_rb,    B), B, 0, stream>>>(ratedby_dst, cntC, E_rb);
  {
    int nbA = cdiv(Mi, 256), nbC = cdiv(Mu, 256);
    rgcn_scan1_kernel<<<nbA, 256, 0, stream>>>(cntA, rpA, bsum, Mi);
    rgcn_scan2_kernel<<<1, 1024, 0, stream>>>(bsum, nbA);
    rgcn_scan3_kernel<<<nbA, 256, 0, stream>>>(rpA, bsum, Mi);
    rgcn_scan1_kernel<<<nbA, 256, 0, stream>>>(cntB, rpB, bsum, Mi);
    rgcn_scan2_kernel<<<1, 1024, 0, stream>>>(bsum, nbA);
    rgcn_scan3_kernel<<<nbA, 256, 0, stream>>>(rpB, bsum, Mi);
    rgcn_scan1_kernel<<<nbC, 256, 0, stream>>>(cntC, rpC, bsum, Mu);
    rgcn_scan2_kernel<<<1, 1024, 0, stream>>>(bsum, nbC);
    rgcn_scan3_kernel<<<nbC, 256, 0, stream>>>(rpC, bsum, Mu);
  }
  rgcn_fill_kernel<<<cdiv(E_rates, B), B, 0, stream>>>(rates_src,   rates_dst,   rpA, curA, colA, E_rates);
  rgcn_fill_kernel<<<cdiv(E_sim,   B), B, 0, stream>>>(sim_src,     sim_dst,     rpB, curB, colB, E_sim);
  rgcn_fill_kernel<<<cdiv(E_rb,    B), B, 0, stream>>>(ratedby_src, ratedby_dst, rpC, curC, colC, E_rb);

  // ---- weight pre-swizzle into WMMA fragment order ----
  rgcn_repack_w_kernel<HID_DIM, IN_DIM><<<cdiv(8192, B), B, 0, stream>>>(W0_rates,   P0_rates);
  rgcn_repack_w_kernel<HID_DIM, IN_DIM><<<cdiv(8192, B), B, 0, stream>>>(W0_sim,     P0_sim);
  rgcn_repack_w_kernel<HID_DIM, IN_DIM><<<cdiv(8192, B), B, 0, stream>>>(W0_ratedby, P0_rb);
  rgcn_repack_w_kernel<OUT_DIM, HID_DIM><<<cdiv(4096, B), B, 0, stream>>>(W1_rates, P1_rates);
  rgcn_repack_w_kernel<OUT_DIM, HID_DIM><<<cdiv(4096, B), B, 0, stream>>>(W1_sim,   P1_sim);
  rgcn_repack_w_kernel<OUT_DIM, HID_DIM><<<cdiv(4096, B), B, 0, stream>>>(W_self,   P_self);

  // ---- combined biases ----
  rgcn_bias_sum_kernel<<<1, HID_DIM, 0, stream>>>(bias_hi, b0_rates, b0_sim, nullptr, HID_DIM);
  rgcn_bias_sum_kernel<<<1, OUT_DIM, 0, stream>>>(out_bias, b_self, b1_rates, b1_sim, OUT_DIM);

  // ---- layer 0 GEMMs (weight-before-aggregate, WMMA) ----
  {
    long wu = cdiv(Mu, 16), wi = cdiv(Mi, 16);
    rgcn_gemm_wmma_kernel<HID_DIM, IN_DIM><<<cdiv(wu * 32, B), B, 0, stream>>>(
        x_user, P0_rates, nullptr, tmp_user, Mu);
    rgcn_gemm_wmma_kernel<HID_DIM, IN_DIM><<<cdiv(wi * 32, B), B, 0, stream>>>(
        x_item, P0_sim, nullptr, tmp_item_sim, Mi);
    rgcn_gemm_wmma_kernel<HID_DIM, IN_DIM><<<cdiv(wi * 32, B), B, 0, stream>>>(
        x_item, P0_rb, nullptr, tmp_item_rb, Mi);
  }

  // ---- layer 0 gather: fused (rates + sim + bias + ReLU) for h_item; (ratedby + bias + ReLU) for h_user ----
  rgcn_gather_kernel<HID_DIM, true, true, false><<<cdiv((long)Mi * 32, B), B, 0, stream>>>(
      tmp_user, rpA, cntA, colA, tmp_item_sim, rpB, cntB, colB, bias_hi, h_item, Mi);
  rgcn_gather_kernel<HID_DIM, false, true, false><<<cdiv((long)Mu * 32, B), B, 0, stream>>>(
      tmp_item_rb, rpC, cntC, colC, nullptr, nullptr, nullptr, nullptr, b0_ratedby, h_user, Mu);

  // ---- layer 1 GEMMs; self-loop GEMM initializes d_out with all biases ----
  {
    long wu = cdiv(Mu, 16), wi = cdiv(Mi, 16);
    rgcn_gemm_wmma_kernel<OUT_DIM, HID_DIM><<<cdiv(wu * 32, B), B, 0, stream>>>(
        h_user, P1_rates, nullptr, tmp2_user, Mu);
    rgcn_gemm_wmma_kernel<OUT_DIM, HID_DIM><<<cdiv(wi * 32, B), B, 0, stream>>>(
        h_item, P1_sim, nullptr, tmp2_item, Mi);
    rgcn_gemm_wmma_kernel<OUT_DIM, HID_DIM><<<cdiv(wi * 32, B), B, 0, stream>>>(
        h_item, P_self, out_bias, out, Mi);
  }

  // ---- layer 1 gather accumulates both etypes onto d_out ----
  rgcn_gather_kernel<OUT_DIM, true, false, true><<<cdiv((long)Mi * 32, B), B, 0, stream>>>(
      tmp2_user, rpA, cntA, colA, tmp2_item, rpB, cntB, colB, nullptr, out, Mi);
}